// HeteroGATLayerA2C_52166672777262
// MI455X (gfx1250) — compile-verified
//
#include <hip/hip_runtime.h>
#include <math.h>

typedef float v2f __attribute__((ext_vector_type(2)));
typedef float v8f __attribute__((ext_vector_type(8)));

#define LRELU(x) ((x) > 0.0f ? (x) : 0.2f * (x))

// ---------------------------------------------------------------------------
// float atomic max via int/uint ordering trick (init buffer to -inf)
// ---------------------------------------------------------------------------
__device__ __forceinline__ void atomicMaxFloat(float* addr, float val) {
    if (val >= 0.0f)
        atomicMax((int*)addr, __float_as_int(val));
    else
        atomicMin((unsigned int*)addr, (unsigned int)__float_as_int(val));
}

// ---------------------------------------------------------------------------
// Fill
// ---------------------------------------------------------------------------
__global__ void fill_k(float* p, float v, size_t n) {
    size_t i = (size_t)blockIdx.x * blockDim.x + threadIdx.x;
    if (i < n) p[i] = v;
}

// ---------------------------------------------------------------------------
// C[M x N] = A[M x 32] @ W[32 x N] + bias  via V_WMMA_F32_16X16X4_F32.
// One wave per 16 x (16*NTILES) strip: A frag loaded once per k-step and fed
// to NTILES chained WMMAs; full tiles take the unguarded store path.
// A-frag layout (ISA 7.12.2): lane<16 -> K={0,1}, lane>=16 -> K={2,3}, M=lane%16
// B-frag: VGPR j holds K = j + 2*(lane/16), N = lane%16
// C/D:    VGPR v holds M = v + 8*(lane/16), N = lane%16
// ---------------------------------------------------------------------------
template <int NTILES>
__global__ void gemm32_wmma(const float* __restrict__ A, const float* __restrict__ W,
                            const float* __restrict__ bias, float* __restrict__ C,
                            int M, int mtiles) {
    constexpr int N = NTILES * 16;
    int mt = blockIdx.x * (blockDim.x >> 5) + (threadIdx.x >> 5);
    if (mt >= mtiles) return;              // wave-uniform exit: EXEC stays full
    int lane = threadIdx.x & 31;
    int half = lane >> 4;
    int l16  = lane & 15;

    int row0 = mt * 16;
    int ar = row0 + l16;
    if (ar >= M) ar = M - 1;               // clamp pad rows (stores handled below)

    v8f acc[NTILES];
    #pragma unroll
    for (int nt = 0; nt < NTILES; ++nt)
        acc[nt] = (v8f){0.f, 0.f, 0.f, 0.f, 0.f, 0.f, 0.f, 0.f};

    #pragma unroll
    for (int ks = 0; ks < 8; ++ks) {
        int k = ks * 4 + half * 2;
        v2f a = *(const v2f*)(A + (size_t)ar * 32 + k);
        #pragma unroll
        for (int nt = 0; nt < NTILES; ++nt) {
            v2f b;
            b.x = W[(size_t)(k + 0) * N + nt * 16 + l16];
            b.y = W[(size_t)(k + 1) * N + nt * 16 + l16];
            acc[nt] = __builtin_amdgcn_wmma_f32_16x16x4_f32(
                false, a, false, b, (short)0, acc[nt], false, false);
        }
    }

    if (row0 + 16 <= M) {                  // full tile: straight-line stores
        #pragma unroll
        for (int nt = 0; nt < NTILES; ++nt) {
            float bi = bias[nt * 16 + l16];
            #pragma unroll
            for (int v = 0; v < 8; ++v) {
                int m = row0 + half * 8 + v;
                C[(size_t)m * N + nt * 16 + l16] = acc[nt][v] + bi;
            }
        }
    } else {                               // tail tile (rare): guarded stores
        #pragma unroll
        for (int nt = 0; nt < NTILES; ++nt) {
            float bi = bias[nt * 16 + l16];
            #pragma unroll
            for (int v = 0; v < 8; ++v) {
                int m = row0 + half * 8 + v;
                if (m < M) C[(size_t)m * N + nt * 16 + l16] = acc[nt][v] + bi;
            }
        }
    }
}

// ---------------------------------------------------------------------------
// Wh_in = hS[1x32] @ W_in[32x64] + b_in   (single row, trivial)
// ---------------------------------------------------------------------------
__global__ void whin_k(const float* __restrict__ hS, const float* __restrict__ W,
                       const float* __restrict__ b, float* __restrict__ out) {
    int c = threadIdx.x;                   // 64 threads
    float s = b[c];
    #pragma unroll
    for (int k = 0; k < 32; ++k) s += hS[k] * W[k * 64 + c];
    out[c] = s;
}

// ---------------------------------------------------------------------------
// Gumbel-hard binarization (forward value == y_hard): msg = (l1+g1 > l0+g0)
// ---------------------------------------------------------------------------
__global__ void discretize_k(float* __restrict__ enc, const float* __restrict__ u,
                             const float* __restrict__ w_bin, const float* __restrict__ b_bin,
                             size_t n) {
    size_t i = (size_t)blockIdx.x * blockDim.x + threadIdx.x;
    if (i >= n) return;
    const float EPS = 1e-10f;
    float x  = enc[i];
    float l0 = x * w_bin[0] + b_bin[0];
    float l1 = x * w_bin[1] + b_bin[1];
    float g0 = -__logf(-__logf(u[2 * i + 0] + EPS) + EPS);
    float g1 = -__logf(-__logf(u[2 * i + 1] + EPS) + EPS);
    enc[i] = (l1 + g1 > l0 + g0) ? 1.0f : 0.0f;
}

// ---------------------------------------------------------------------------
// out[i] = dot(z[i, 0:64], a[aoff : aoff+64])  -- one wave per row
// ---------------------------------------------------------------------------
__global__ void rowdot_k(const float* __restrict__ z, const float* __restrict__ a,
                         int aoff, float* __restrict__ out, int n) {
    int row  = blockIdx.x * (blockDim.x >> 5) + (threadIdx.x >> 5);
    if (row >= n) return;
    int lane = threadIdx.x & 31;
    float s = z[(size_t)row * 64 + lane]      * a[aoff + lane]
            + z[(size_t)row * 64 + 32 + lane] * a[aoff + 32 + lane];
    #pragma unroll
    for (int m = 16; m > 0; m >>= 1) s += __shfl_xor(s, m, 32);
    if (lane == 0) out[row] = s;
}

// scalar dot of one 64-vector with a[aoff:] -> out[0]   (launch <<<1,32>>>)
__global__ void sdot_k(const float* __restrict__ x, const float* __restrict__ a,
                       int aoff, float* __restrict__ out) {
    int lane = threadIdx.x;
    float s = x[lane] * a[aoff + lane] + x[lane + 32] * a[aoff + lane + 32];
    #pragma unroll
    for (int m = 16; m > 0; m >>= 1) s += __shfl_xor(s, m, 32);
    if (lane == 0) out[0] = s;
}

// ---------------------------------------------------------------------------
// Edge passes (per relation)
// ---------------------------------------------------------------------------
__global__ void edge_max_k(const int* __restrict__ src, const int* __restrict__ dst,
                           const float* __restrict__ esrc, const float* __restrict__ edst,
                           float* __restrict__ ebuf, float* __restrict__ maxb, int E) {
    int e = blockIdx.x * blockDim.x + threadIdx.x;
    if (e >= E) return;
    float v = esrc[src[e]] + edst[dst[e]];
    v = LRELU(v);
    ebuf[e] = v;
    atomicMaxFloat(&maxb[dst[e]], v);
}

__global__ void edge_den_k(const int* __restrict__ dst, float* __restrict__ ebuf,
                           const float* __restrict__ maxb, float* __restrict__ den, int E) {
    int e = blockIdx.x * blockDim.x + threadIdx.x;
    if (e >= E) return;
    int d = dst[e];
    float ex = __expf(ebuf[e] - maxb[d]);
    ebuf[e] = ex;
    atomicAdd(&den[d], ex);
}

// 16 threads per edge, 4 columns each; weight folded with 1/den so both
// relations into a node type share one recv buffer.
__global__ void edge_acc_k(const int* __restrict__ src, const int* __restrict__ dst,
                           const float* __restrict__ ebuf, const float* __restrict__ den,
                           const float* __restrict__ z, float* __restrict__ recv, int E) {
    int t = blockIdx.x * blockDim.x + threadIdx.x;
    int e = t >> 4;
    if (e >= E) return;
    int c0 = (t & 15) * 4;
    int s  = src[e], d = dst[e];
    float w = ebuf[e] / den[d];
    float4 zz = *(const float4*)(z + (size_t)s * 64 + c0);
    float* o = recv + (size_t)d * 64 + c0;
    atomicAdd(o + 0, w * zz.x);
    atomicAdd(o + 1, w * zz.y);
    atomicAdd(o + 2, w * zz.z);
    atomicAdd(o + 3, w * zz.w);
}

// ---------------------------------------------------------------------------
// All-nodes -> single state node relations (ps / as): block reductions
// ---------------------------------------------------------------------------
__global__ void s_max_k(const float* __restrict__ esrc, const float* __restrict__ edstp,
                        float* __restrict__ maxb, int n) {
    __shared__ float sm[256];
    int i = blockIdx.x * 256 + threadIdx.x;
    float v = -INFINITY;
    if (i < n) { float e = esrc[i] + edstp[0]; v = LRELU(e); }
    sm[threadIdx.x] = v;
    __syncthreads();
    for (int s = 128; s > 0; s >>= 1) {
        if (threadIdx.x < s) sm[threadIdx.x] = fmaxf(sm[threadIdx.x], sm[threadIdx.x + s]);
        __syncthreads();
    }
    if (threadIdx.x == 0) atomicMaxFloat(maxb, sm[0]);
}

__global__ void s_den_k(const float* __restrict__ esrc, const float* __restrict__ edstp,
                        const float* __restrict__ maxb, float* __restrict__ den, int n) {
    __shared__ float sm[256];
    int i = blockIdx.x * 256 + threadIdx.x;
    float v = 0.0f;
    if (i < n) { float e = esrc[i] + edstp[0]; e = LRELU(e); v = __expf(e - maxb[0]); }
    sm[threadIdx.x] = v;
    __syncthreads();
    for (int s = 128; s > 0; s >>= 1) {
        if (threadIdx.x < s) sm[threadIdx.x] += sm[threadIdx.x + s];
        __syncthreads();
    }
    if (threadIdx.x == 0) atomicAdd(den, sm[0]);
}

__global__ void s_acc_k(const float* __restrict__ esrc, const float* __restrict__ edstp,
                        const float* __restrict__ maxb, const float* __restrict__ den,
                        const float* __restrict__ z, float* __restrict__ recvS, int n) {
    __shared__ float sacc[64];
    if (threadIdx.x < 64) sacc[threadIdx.x] = 0.0f;
    __syncthreads();
    int i = blockIdx.x * 256 + threadIdx.x;
    if (i < n) {
        float e = esrc[i] + edstp[0];
        e = LRELU(e);
        float w = __expf(e - maxb[0]) / den[0];
        #pragma unroll 4
        for (int c = 0; c < 64; ++c) atomicAdd(&sacc[c], w * z[(size_t)i * 64 + c]);
    }
    __syncthreads();
    if (threadIdx.x < 64) atomicAdd(&recvS[threadIdx.x], sacc[threadIdx.x]);
}

// ---------------------------------------------------------------------------
// out = relu(Wh + recv)
// ---------------------------------------------------------------------------
__global__ void finalize_k(const float* __restrict__ Wh, const float* __restrict__ recv,
                           float* __restrict__ out, size_t n) {
    size_t i = (size_t)blockIdx.x * blockDim.x + threadIdx.x;
    if (i >= n) return;
    float v = Wh[i] + recv[i];
    out[i] = v > 0.0f ? v : 0.0f;
}

// ---------------------------------------------------------------------------
// Host driver
// ---------------------------------------------------------------------------
static inline unsigned cdiv(size_t a, unsigned b) { return (unsigned)((a + b - 1) / b); }

extern "C" void kernel_launch(void* const* d_in, const int* in_sizes, int n_in,
                              void* d_out, int out_size, void* d_ws, size_t ws_size,
                              hipStream_t stream) {
    const float* hP = (const float*)d_in[0];
    const float* hA = (const float*)d_in[1];
    const float* hS = (const float*)d_in[2];
    const float* uP = (const float*)d_in[3];
    const float* uA = (const float*)d_in[4];
    const int* pp_src = (const int*)d_in[5];
    const int* pp_dst = (const int*)d_in[6];
    const int* pa_src = (const int*)d_in[7];
    const int* pa_dst = (const int*)d_in[8];
    const int* ap_src = (const int*)d_in[9];
    const int* ap_dst = (const int*)d_in[10];
    const int* aa_src = (const int*)d_in[11];
    const int* aa_dst = (const int*)d_in[12];
    // 13..18: ps/as/in edge lists are identity/zeros; handled analytically.
    const float* W_P   = (const float*)d_in[19]; const float* b_P   = (const float*)d_in[20];
    const float* W_A   = (const float*)d_in[21]; const float* b_A   = (const float*)d_in[22];
    const float* W_p2s = (const float*)d_in[23]; const float* b_p2s = (const float*)d_in[24];
    const float* W_a2s = (const float*)d_in[25]; const float* b_a2s = (const float*)d_in[26];
    const float* W_in  = (const float*)d_in[27]; const float* b_in  = (const float*)d_in[28];
    const float* We_P  = (const float*)d_in[29]; const float* be_P  = (const float*)d_in[30];
    const float* We_A  = (const float*)d_in[31]; const float* be_A  = (const float*)d_in[32];
    const float* w_bin = (const float*)d_in[33]; const float* b_bin = (const float*)d_in[34];
    const float* Wd_P  = (const float*)d_in[35]; const float* bd_P  = (const float*)d_in[36];
    const float* Wd_A  = (const float*)d_in[37]; const float* bd_A  = (const float*)d_in[38];
    const float* a_pp  = (const float*)d_in[39];
    const float* a_pa  = (const float*)d_in[40];
    const float* a_ap  = (const float*)d_in[41];
    const float* a_aa  = (const float*)d_in[42];
    const float* a_ps  = (const float*)d_in[43];
    const float* a_as  = (const float*)d_in[44];

    const int NP = in_sizes[0] / 32;
    const int NA = in_sizes[1] / 32;
    const int E_PP = in_sizes[5], E_PA = in_sizes[7], E_AP = in_sizes[9], E_AA = in_sizes[11];

    // ---- workspace carve-out (floats, 16-float aligned blocks) ----
    float* ws = (float*)d_ws;
    size_t off = 0;
    auto F = [&](size_t n) -> float* {
        float* p = ws + off;
        off += (n + 15) & ~(size_t)15;
        return p;
    };
    float* WhP   = F((size_t)NP * 64);
    float* WhA   = F((size_t)NA * 64);
    float* Whp2s = F((size_t)NP * 64);
    float* Wha2s = F((size_t)NA * 64);
    float* WhIn  = F(64);
    float* msgP  = F((size_t)NP * 32);   // enc_P then msg_P in place
    float* msgA  = F((size_t)NA * 32);
    float* mPP   = F((size_t)NP * 64);   // msg_P @ Wd_P
    float* mPA   = F((size_t)NP * 64);   // msg_P @ Wd_A
    float* mAP   = F((size_t)NA * 64);   // msg_A @ Wd_P
    float* mAA   = F((size_t)NA * 64);   // msg_A @ Wd_A
    float* esrc_pp = F(NP); float* esrc_pa = F(NP); float* esrc_ps = F(NP);
    float* esrc_ap = F(NA); float* esrc_aa = F(NA); float* esrc_as = F(NA);
    float* edst_pp = F(NP); float* edst_ap = F(NP);
    float* edst_pa = F(NA); float* edst_aa = F(NA);
    float* edst_ps = F(16); float* edst_as = F(16);
    float* max_reg = ws + off;           // contiguous -inf region start
    float* max_pp = F(NP); float* max_ap = F(NP);
    float* max_pa = F(NA); float* max_aa = F(NA);
    float* max_ps = F(16); float* max_as = F(16);
    float* zero_reg = ws + off;          // contiguous zero region start
    float* den_pp = F(NP); float* den_ap = F(NP);
    float* den_pa = F(NA); float* den_aa = F(NA);
    float* den_ps = F(16); float* den_as = F(16);
    float* recvP = F((size_t)NP * 64);
    float* recvA = F((size_t)NA * 64);
    float* recvS = F(64);
    float* zero_end = ws + off;
    float* ebuf = F((size_t)(E_PP > E_AA ? E_PP : E_AA));
    if (off * sizeof(float) > ws_size) return;  // insufficient scratch

    // ---- per-call init (graph-replay safe) ----
    {
        size_t nmax = (size_t)(zero_reg - max_reg);
        fill_k<<<cdiv(nmax, 256), 256, 0, stream>>>(max_reg, -INFINITY, nmax);
        size_t nzero = (size_t)(zero_end - zero_reg);
        fill_k<<<cdiv(nzero, 256), 256, 0, stream>>>(zero_reg, 0.0f, nzero);
    }

    // ---- dense projections (WMMA f32, 16x64 / 16x32 strips per wave) ----
    auto gemm64 = [&](const float* A, const float* W, const float* b, float* C, int M) {
        int mt = (M + 15) / 16;
        gemm32_wmma<4><<<cdiv(mt, 8), 256, 0, stream>>>(A, W, b, C, M, mt);
    };
    auto gemm32n = [&](const float* A, const float* W, const float* b, float* C, int M) {
        int mt = (M + 15) / 16;
        gemm32_wmma<2><<<cdiv(mt, 8), 256, 0, stream>>>(A, W, b, C, M, mt);
    };
    gemm64(hP, W_P,   b_P,   WhP,   NP);
    gemm64(hA, W_A,   b_A,   WhA,   NA);
    gemm64(hP, W_p2s, b_p2s, Whp2s, NP);
    gemm64(hA, W_a2s, b_a2s, Wha2s, NA);
    gemm32n(hP, We_P, be_P,  msgP,  NP);
    gemm32n(hA, We_A, be_A,  msgA,  NA);
    whin_k<<<1, 64, 0, stream>>>(hS, W_in, b_in, WhIn);

    // ---- binarize (in place) ----
    discretize_k<<<cdiv((size_t)NP * 32, 256), 256, 0, stream>>>(msgP, uP, w_bin, b_bin, (size_t)NP * 32);
    discretize_k<<<cdiv((size_t)NA * 32, 256), 256, 0, stream>>>(msgA, uA, w_bin, b_bin, (size_t)NA * 32);

    // ---- decoders ----
    gemm64(msgP, Wd_P, bd_P, mPP, NP);
    gemm64(msgP, Wd_A, bd_A, mPA, NP);
    gemm64(msgA, Wd_P, bd_P, mAP, NA);
    gemm64(msgA, Wd_A, bd_A, mAA, NA);

    // ---- attention logits: per-node projections ----
    auto rdot = [&](const float* z, const float* a, int aoff, float* o, int n) {
        rowdot_k<<<cdiv(n, 8), 256, 0, stream>>>(z, a, aoff, o, n);
    };
    rdot(mPP,   a_pp, 0,  esrc_pp, NP);
    rdot(mPA,   a_pa, 0,  esrc_pa, NP);
    rdot(Whp2s, a_ps, 0,  esrc_ps, NP);
    rdot(mAP,   a_ap, 0,  esrc_ap, NA);
    rdot(mAA,   a_aa, 0,  esrc_aa, NA);
    rdot(Wha2s, a_as, 0,  esrc_as, NA);
    rdot(WhP,   a_pp, 64, edst_pp, NP);
    rdot(WhP,   a_ap, 64, edst_ap, NP);
    rdot(WhA,   a_pa, 64, edst_pa, NA);
    rdot(WhA,   a_aa, 64, edst_aa, NA);
    sdot_k<<<1, 32, 0, stream>>>(WhIn, a_ps, 64, edst_ps);
    sdot_k<<<1, 32, 0, stream>>>(WhIn, a_as, 64, edst_as);

    // ---- per-relation segment softmax + weighted scatter ----
    auto relation = [&](const int* src, const int* dst, int E,
                        const float* esrc, const float* edst,
                        float* maxb, float* den, const float* z, float* recv) {
        edge_max_k<<<cdiv(E, 256), 256, 0, stream>>>(src, dst, esrc, edst, ebuf, maxb, E);
        edge_den_k<<<cdiv(E, 256), 256, 0, stream>>>(dst, ebuf, maxb, den, E);
        edge_acc_k<<<cdiv((size_t)E * 16, 256), 256, 0, stream>>>(src, dst, ebuf, den, z, recv, E);
    };
    relation(pp_src, pp_dst, E_PP, esrc_pp, edst_pp, max_pp, den_pp, mPP, recvP);
    relation(ap_src, ap_dst, E_AP, esrc_ap, edst_ap, max_ap, den_ap, mAP, recvP);
    relation(pa_src, pa_dst, E_PA, esrc_pa, edst_pa, max_pa, den_pa, mPA, recvA);
    relation(aa_src, aa_dst, E_AA, esrc_aa, edst_aa, max_aa, den_aa, mAA, recvA);

    // ---- state-node relations (all-to-one) ----
    auto srel = [&](const float* esrc, const float* edstp, float* maxb, float* den,
                    const float* z, int n) {
        s_max_k<<<cdiv(n, 256), 256, 0, stream>>>(esrc, edstp, maxb, n);
        s_den_k<<<cdiv(n, 256), 256, 0, stream>>>(esrc, edstp, maxb, den, n);
        s_acc_k<<<cdiv(n, 256), 256, 0, stream>>>(esrc, edstp, maxb, den, z, recvS, n);
    };
    srel(esrc_ps, edst_ps, max_ps, den_ps, Whp2s, NP);
    srel(esrc_as, edst_as, max_as, den_as, Wha2s, NA);

    // ---- outputs: [out_P | out_A | out_S] ----
    float* out = (float*)d_out;
    finalize_k<<<cdiv((size_t)NP * 64, 256), 256, 0, stream>>>(WhP, recvP, out, (size_t)NP * 64);
    finalize_k<<<cdiv((size_t)NA * 64, 256), 256, 0, stream>>>(WhA, recvA, out + (size_t)NP * 64, (size_t)NA * 64);
    // out_S = relu(recvS + Wh_in)   (self-loop segment_sum contributes Wh_in)
    finalize_k<<<1, 64, 0, stream>>>(WhIn, recvS, out + (size_t)(NP + NA) * 64, 64);
}